// ThresholdCanny_70798240907723
// MI455X (gfx1250) — compile-verified
//
#include <hip/hip_runtime.h>
#include <cstddef>

// ---------------------------------------------------------------------------
// Canny (ThresholdCanny) for MI455X / gfx1250.
// Bandwidth-bound pipeline (~650MB traffic -> ~28us at 23.3TB/s); whole
// intermediate working set (~180MB) is L2-resident (192MB).
// Gaussian row pass uses V_WMMA_F32_16X16X4_F32 (exact fp32 matrix path).
// Filter taps are inline literals selected with v_cndmask (no constant-mem
// loads in the WMMA inner loop).
// ---------------------------------------------------------------------------

typedef float v2f __attribute__((ext_vector_type(2)));
typedef float v8f __attribute__((ext_vector_type(8)));

#define IMG_H 1024
#define IMG_W 1024

// normalized 1-D Gaussian, size 5, sigma 1.0 (symmetric: w0=w4, w1=w3)
#define GW0 0.054488685f
#define GW1 0.244201342f
#define GW2 0.402619947f

// Branchless banded-matrix weight: t in [0,4] -> tap, else 0.
__device__ __forceinline__ float wsel(int t) {
  float r = 0.0f;
  r = (t == 0 || t == 4) ? GW0 : r;
  r = (t == 1 || t == 3) ? GW1 : r;
  r = (t == 2)           ? GW2 : r;
  return r;
}

// ---------------------------------------------------------------------------
// K1: 5x5 Gaussian blur of green channel, one 16x16 tile per wave.
// Row pass via WMMA f32 16x16x4 (banded-matrix product), column pass in VALU.
// ---------------------------------------------------------------------------
__global__ __launch_bounds__(32) void k_blur_wmma(const float* __restrict__ x,
                                                  float* __restrict__ blur) {
  const int lane = threadIdx.x;
  const int t    = blockIdx.x;
  const int tx   = t & 63;
  const int ty   = (t >> 6) & 63;
  const int n    = t >> 12;
  const float* g = x + ((size_t)n * 3 + 1) * (size_t)(IMG_H * IMG_W);  // green
  const int r0 = ty * 16, c0 = tx * 16;

  __shared__ float tin[20][21];  // input tile with halo 2 (zero 'SAME' pad)
  __shared__ float rc[20][17];   // row-convolved band

  for (int i = lane; i < 400; i += 32) {
    int r = i / 20, c = i - r * 20;
    int gr = r0 - 2 + r, gc = c0 - 2 + c;
    float v = 0.0f;
    if (gr >= 0 && gr < IMG_H && gc >= 0 && gc < IMG_W)
      v = g[(size_t)gr * IMG_W + gc];
    tin[r][c] = v;
  }
  __syncthreads();

  const int m    = lane & 15;   // A: M row / B: N col / D: N col
  const int half = lane >> 4;   // K-half / M-half selector
  const int koff = half * 2;

  // B-matrix (banded Gaussian) is the same for both row bands: precompute the
  // 5 chunks x 2 VGPRs of per-lane weights branchlessly in registers.
  v2f bw[5];
#pragma unroll
  for (int c = 0; c < 5; ++c) {
    int k0 = c * 4 + koff;
    bw[c].x = wsel(k0 - m);
    bw[c].y = wsel(k0 + 1 - m);
  }

  // Row conv: RC[i][c] = sum_k tin[i][k] * W[k][c], W[k][c] = G5[k-c] (banded)
  for (int pass = 0; pass < 2; ++pass) {
    const int rb = pass * 4;    // band rows rb..rb+15
    v8f acc = {};
#pragma unroll
    for (int c = 0; c < 5; ++c) {
      const int kb = c * 4;
      v2f a;
      a.x = tin[rb + m][kb + koff];
      a.y = tin[rb + m][kb + koff + 1];
      acc = __builtin_amdgcn_wmma_f32_16x16x4_f32(false, a, false, bw[c],
                                                  (short)0, acc, false, false);
    }
#pragma unroll
    for (int v = 0; v < 8; ++v) {
      int row = rb + v + half * 8;            // D layout: rows v + half*8
      if (pass == 0 || row >= 16) rc[row][m] = acc[v];
    }
  }
  __syncthreads();

  // Column pass: out[r][c] = GW0*(RC[r]+RC[r+4]) + GW1*(RC[r+1]+RC[r+3]) + GW2*RC[r+2]
#pragma unroll
  for (int v = 0; v < 8; ++v) {
    int r = v + half * 8;
    float o = GW0 * (rc[r][m] + rc[r + 4][m]) +
              GW1 * (rc[r + 1][m] + rc[r + 3][m]) +
              GW2 * rc[r + 2][m];
    blur[(size_t)n * IMG_H * IMG_W + (size_t)(r0 + r) * IMG_W + (c0 + m)] = o;
  }
}

// ---------------------------------------------------------------------------
// K2: Sobel -> magnitude + direction class; per-image max via atomics.
// ---------------------------------------------------------------------------
__global__ __launch_bounds__(256) void k_grad(const float* __restrict__ blur,
                                              float* __restrict__ mag,
                                              unsigned char* __restrict__ dir,
                                              unsigned* __restrict__ maxbuf) {
  const size_t idx = (size_t)blockIdx.x * 256 + threadIdx.x;
  const int n = (int)(idx >> 20);
  const int y = (int)(idx >> 10) & 1023;
  const int x = (int)idx & 1023;
  const float* b = blur + (size_t)n * IMG_H * IMG_W;

  auto ld = [&](int yy, int xx) -> float {
    return (yy >= 0 && yy < IMG_H && xx >= 0 && xx < IMG_W)
               ? b[(size_t)yy * IMG_W + xx] : 0.0f;
  };
  float a00 = ld(y - 1, x - 1), a01 = ld(y - 1, x), a02 = ld(y - 1, x + 1);
  float a10 = ld(y, x - 1),                          a12 = ld(y, x + 1);
  float a20 = ld(y + 1, x - 1), a21 = ld(y + 1, x), a22 = ld(y + 1, x + 1);

  float gx = (a02 - a00) + 2.0f * (a12 - a10) + (a22 - a20);
  float gy = (a20 - a00) + 2.0f * (a21 - a01) + (a22 - a02);
  float mg = sqrtf(gx * gx + gy * gy + 1e-12f);

  // Direction class without atan2 (flip-invariant): 0:0deg 1:45 2:90 3:135
  float ax = fabsf(gx), ay = fabsf(gy);
  unsigned char d;
  if (ay < 0.41421356f * ax)        d = 0;
  else if (ay >= 2.41421356f * ax)  d = 2;
  else                              d = (gx * gy >= 0.0f) ? 1 : 3;

  mag[idx] = mg;
  dir[idx] = d;

  __shared__ unsigned smax;
  if (threadIdx.x == 0) smax = 0u;
  __syncthreads();
  atomicMax(&smax, __float_as_uint(mg));  // mg > 0 -> uint order == float order
  __syncthreads();
  if (threadIdx.x == 0) atomicMax(&maxbuf[n], smax);
}

// ---------------------------------------------------------------------------
// K3: NMS + double threshold -> state {0:none,1:weak,2:strong}
// ---------------------------------------------------------------------------
__global__ __launch_bounds__(256) void k_nms(const float* __restrict__ mag,
                                             const unsigned char* __restrict__ dir,
                                             const unsigned* __restrict__ maxbuf,
                                             const float* __restrict__ lowp,
                                             const float* __restrict__ highp,
                                             unsigned char* __restrict__ st) {
  const size_t idx = (size_t)blockIdx.x * 256 + threadIdx.x;
  const int n = (int)(idx >> 20);
  const int y = (int)(idx >> 10) & 1023;
  const int x = (int)idx & 1023;

  const float M = __uint_as_float(maxbuf[n]);
  const float denom = M + 1e-12f;
  const float tl = lowp[0] * denom;   // nms/denom >= low  <=>  nms >= tl
  const float th = highp[0] * denom;

  const float* mg = mag + (size_t)n * IMG_H * IMG_W;
  auto ld = [&](int yy, int xx) -> float {
    return (yy >= 0 && yy < IMG_H && xx >= 0 && xx < IMG_W)
               ? mg[(size_t)yy * IMG_W + xx] : 0.0f;
  };
  const float m0 = mg[(size_t)y * IMG_W + x];
  const int d = dir[idx];
  int dy1, dx1, dy2, dx2;
  if (d == 0)      { dy1 = 0;  dx1 = 1;  dy2 = 0; dx2 = -1; }
  else if (d == 1) { dy1 = -1; dx1 = 1;  dy2 = 1; dx2 = -1; }
  else if (d == 2) { dy1 = -1; dx1 = 0;  dy2 = 1; dx2 = 0;  }
  else             { dy1 = -1; dx1 = -1; dy2 = 1; dx2 = 1;  }
  const float n1 = ld(y + dy1, x + dx1);
  const float n2 = ld(y + dy2, x + dx2);
  const float nms = (m0 >= n1 && m0 >= n2) ? m0 : 0.0f;
  st[idx] = (nms >= th) ? 2 : ((nms >= tl) ? 1 : 0);
}

// ---------------------------------------------------------------------------
// K4: hysteresis pass -- 32x32 tile, block-local monotone fixpoint in LDS.
// Deterministic: only 1->2 transitions; loop exits at tile-local fixpoint.
// ---------------------------------------------------------------------------
__global__ __launch_bounds__(256) void k_hyst(const unsigned char* __restrict__ cur,
                                              unsigned char* __restrict__ nxt) {
  const int t  = blockIdx.x;
  const int tx = t & 31;
  const int ty = (t >> 5) & 31;
  const int n  = t >> 10;
  const int x0 = tx * 32, y0 = ty * 32;
  const unsigned char* c = cur + (size_t)n * IMG_H * IMG_W;

  __shared__ unsigned char st[34][36];
  __shared__ int changed;

  for (int p = threadIdx.x; p < 34 * 34; p += 256) {
    int ly = p / 34, lx = p - ly * 34;
    int gy = y0 - 1 + ly, gx = x0 - 1 + lx;
    unsigned char v = 0;
    if (gy >= 0 && gy < IMG_H && gx >= 0 && gx < IMG_W)
      v = c[(size_t)gy * IMG_W + gx];
    st[ly][lx] = v;
  }
  __syncthreads();

  for (int it = 0; it < 96; ++it) {
    if (threadIdx.x == 0) changed = 0;
    __syncthreads();
    int lch = 0;
#pragma unroll
    for (int k = 0; k < 4; ++k) {
      int p = threadIdx.x + k * 256;
      int ly = (p >> 5) + 1, lx = (p & 31) + 1;
      if (st[ly][lx] == 1) {
        int e = st[ly-1][lx-1] | st[ly-1][lx] | st[ly-1][lx+1] |
                st[ly  ][lx-1] |                st[ly  ][lx+1] |
                st[ly+1][lx-1] | st[ly+1][lx] | st[ly+1][lx+1];
        if (e & 2) { st[ly][lx] = 2; lch = 1; }
      }
    }
    if (lch) changed = 1;
    __syncthreads();
    if (!changed) break;
  }

  unsigned char* o = nxt + (size_t)n * IMG_H * IMG_W;
  for (int k = 0; k < 4; ++k) {
    int p = threadIdx.x + k * 256;
    int ly = p >> 5, lx = p & 31;
    o[(size_t)(y0 + ly) * IMG_W + (x0 + lx)] = st[ly + 1][lx + 1];
  }
}

// ---------------------------------------------------------------------------
// K5: out = 50 * (edges - threshold)
// ---------------------------------------------------------------------------
__global__ __launch_bounds__(256) void k_final(const unsigned char* __restrict__ st,
                                               const float* __restrict__ thr,
                                               float* __restrict__ out) {
  const size_t idx = (size_t)blockIdx.x * 256 + threadIdx.x;
  out[idx] = 50.0f * ((st[idx] == 2 ? 1.0f : 0.0f) - thr[0]);
}

// ---------------------------------------------------------------------------
extern "C" void kernel_launch(void* const* d_in, const int* in_sizes, int n_in,
                              void* d_out, int out_size, void* d_ws, size_t ws_size,
                              hipStream_t stream) {
  const float* x    = (const float*)d_in[0];
  const float* thr  = (const float*)d_in[1];
  const float* low  = (const float*)d_in[2];
  const float* high = (const float*)d_in[3];
  float* out = (float*)d_out;

  const int N = in_sizes[0] / (3 * IMG_H * IMG_W);
  const size_t HW = (size_t)IMG_H * IMG_W;
  const size_t NP = (size_t)N * HW;

  // Workspace layout (~176 MiB for N=16): all fully rewritten each call.
  float* blur        = (float*)d_ws;
  float* mag         = blur + NP;
  unsigned char* dir = (unsigned char*)(mag + NP);
  unsigned char* stA = dir + NP;
  unsigned char* stB = stA + NP;
  unsigned* maxbuf   = (unsigned*)(stB + NP);

  hipMemsetAsync(maxbuf, 0, (size_t)N * sizeof(unsigned), stream);

  const int pixblocks = (int)(NP / 256);
  k_blur_wmma<<<N * 64 * 64, dim3(32), 0, stream>>>(x, blur);
  k_grad<<<pixblocks, dim3(256), 0, stream>>>(blur, mag, dir, maxbuf);
  k_nms<<<pixblocks, dim3(256), 0, stream>>>(mag, dir, maxbuf, low, high, stA);

  unsigned char* cur = stA;
  unsigned char* nxt = stB;
  for (int p = 0; p < 10; ++p) {  // even count -> result lands back in stA
    k_hyst<<<N * 32 * 32, dim3(256), 0, stream>>>(cur, nxt);
    unsigned char* tmp = cur; cur = nxt; nxt = tmp;
  }
  k_final<<<pixblocks, dim3(256), 0, stream>>>(cur, thr, out);
}